// OFFLINE_AE_RPN_48619029791352
// MI455X (gfx1250) — compile-verified
//
#include <hip/hip_runtime.h>
#include <hip/hip_bf16.h>
#include <math.h>

// ---------------- problem constants (from reference) ----------------
#define NN 8
#define CC 256
#define HH 128
#define WW 128
#define GG 64
#define AA 3
#define DD 64
#define MM (HH * WW)          // 16384 locations per image
#define NMTOT (NN * MM)       // 131072 pixels total
// total err elements = 8*256*128*128 = 2^25
#define AE_VEC4 (1 << 23)     // 33554432 / 4 float4 loads

#define AE_BLOCKS 1024
#define AE_THREADS 256
#define DEN_THREADS 256       // 8 waves -> 128 pixels / block
#define DEN_BLOCKS (NMTOT / 128)   // 1024

typedef __attribute__((ext_vector_type(2))) float v2f;
typedef __attribute__((ext_vector_type(8))) float v8f;

// ---------------- kernel 1: loss_ae partial sums (deterministic) ----------------
__global__ void k_ae_partial(const float* __restrict__ err, float* __restrict__ partial) {
    __shared__ float red[AE_THREADS];
    const float4* e4 = (const float4*)err;
    float acc = 0.0f;
    for (int i = blockIdx.x * blockDim.x + threadIdx.x; i < AE_VEC4;
         i += gridDim.x * blockDim.x) {
        float4 v = e4[i];
        acc = fmaf(v.x, v.x, acc);
        acc = fmaf(v.y, v.y, acc);
        acc = fmaf(v.z, v.z, acc);
        acc = fmaf(v.w, v.w, acc);
    }
    red[threadIdx.x] = acc;
    __syncthreads();
    for (int s = AE_THREADS / 2; s > 0; s >>= 1) {
        if (threadIdx.x < s) red[threadIdx.x] += red[threadIdx.x + s];
        __syncthreads();
    }
    if (threadIdx.x == 0) partial[blockIdx.x] = red[0];
}

// ---------------- kernel 2: anchor matching -> unknown-class mask ----------------
__global__ void k_mask(const float* __restrict__ gt_boxes, const int* __restrict__ gt_classes,
                       float* __restrict__ mask) {
    int idx = blockIdx.x * blockDim.x + threadIdx.x;
    if (idx >= NMTOT) return;
    int n  = idx >> 14;           // / MM
    int m  = idx & (MM - 1);
    int ix = m & (WW - 1);
    int iy = m >> 7;
    float sx = (float)ix * 8.0f;  // STRIDE
    float sy = (float)iy * 8.0f;

    const float* gb = gt_boxes + (size_t)n * GG * 4;
    const int*   gc = gt_classes + (size_t)n * GG;

    const float ratios[AA] = {0.5f, 1.0f, 2.0f};
    bool unk = false;
    #pragma unroll
    for (int a = 0; a < AA; ++a) {
        float sr = sqrtf(ratios[a]);
        float wa = 32.0f / sr;
        float ha = 32.0f * sr;
        float ax1 = sx - 0.5f * wa, ay1 = sy - 0.5f * ha;
        float ax2 = sx + 0.5f * wa, ay2 = sy + 0.5f * ha;
        float area_a = wa * ha;
        float best = -1.0f;
        int   bg = 0;
        for (int g = 0; g < GG; ++g) {
            float bx1 = gb[g * 4 + 0], by1 = gb[g * 4 + 1];
            float bx2 = gb[g * 4 + 2], by2 = gb[g * 4 + 3];
            float area_b = (bx2 - bx1) * (by2 - by1);
            float lx = fmaxf(bx1, ax1), ly = fmaxf(by1, ay1);
            float rx = fminf(bx2, ax2), ry = fminf(by2, ay2);
            float iw = fmaxf(rx - lx, 0.0f), ih = fmaxf(ry - ly, 0.0f);
            float inter = iw * ih;
            float iou = inter / fmaxf(area_b + area_a - inter, 1e-9f);
            if (iou > best) { best = iou; bg = g; }   // strict > == argmax first-tie
        }
        unk = unk || (gc[bg] == 80);                  // UNK_CLASS
    }
    mask[idx] = unk ? 1.0f : 0.0f;
}

// ---------------- kernel 3: density MLP via f32 WMMA + BCE partials ----------------
// Each wave handles a 16-pixel tile. Layer2 [16x64]x[64x32] done as 16 K-steps
// of V_WMMA_F32_16X16X4_F32 into two 16x16 accumulators (e=0..15, e=16..31).
__global__ void k_density(const float* __restrict__ err_map,
                          const float* __restrict__ w1, const float* __restrict__ b1,
                          const float* __restrict__ w2, const float* __restrict__ b2,
                          const float* __restrict__ w3, const float* __restrict__ b3,
                          const float* __restrict__ mask,
                          float* __restrict__ partial) {
    __shared__ float wsum[DEN_THREADS / 32];
    const int lane = threadIdx.x & 31;
    const int wv   = threadIdx.x >> 5;
    const int tile = blockIdx.x * (DEN_THREADS / 32) + wv;
    const int base = tile * 16;                 // tiles never cross image boundary (MM % 16 == 0)

    const int grpHi = lane >> 4;                // 0: lanes 0-15, 1: lanes 16-31
    const int l15   = lane & 15;

    // lane's pixel (column index of A row); both half-waves mirror the 16 pixels
    const float pix = err_map[base + l15];
    const float b3v = b3[0];

    v8f c0 = {};                                // D tile, e = 0..15
    v8f c1 = {};                                // D tile, e = 16..31
    #pragma unroll
    for (int k0 = 0; k0 < 16; ++k0) {
        const int d0 = 4 * k0 + 2 * grpHi;      // A: v0/v1 = K {0,1} (lo lanes) / {2,3} (hi lanes)
        v2f a;
        a.x = fmaxf(fmaf(pix, w1[d0 + 0], b1[d0 + 0]), 0.0f);   // h1 = relu(pix*w1+b1)
        a.y = fmaxf(fmaf(pix, w1[d0 + 1], b1[d0 + 1]), 0.0f);
        // B 4x16: lower lanes hold rows K0/K1, upper lanes K2/K3; N = l15
        v2f bl, bh;
        bl.x = w2[(d0 + 0) * 32 + l15];
        bl.y = w2[(d0 + 1) * 32 + l15];
        bh.x = w2[(d0 + 0) * 32 + 16 + l15];
        bh.y = w2[(d0 + 1) * 32 + 16 + l15];
        c0 = __builtin_amdgcn_wmma_f32_16x16x4_f32(false, a, false, bl, (short)0, c0, false, false);
        c1 = __builtin_amdgcn_wmma_f32_16x16x4_f32(false, a, false, bh, (short)0, c1, false, false);
    }

    // epilogue: +b2, relu, dot with w3, 16-lane butterfly, softplus BCE
    const float b2lo = b2[l15];
    const float b2hi = b2[l15 + 16];
    const float w3lo = w3[l15];
    const float w3hi = w3[l15 + 16];
    float acc = 0.0f;
    #pragma unroll
    for (int i = 0; i < 8; ++i) {
        // C/D element (VGPR i, this lane) = (m = i + 8*grpHi, e = 16*nt + l15)
        float h0 = fmaxf(c0[i] + b2lo, 0.0f);
        float h1 = fmaxf(c1[i] + b2hi, 0.0f);
        float t  = fmaf(h0, w3lo, h1 * w3hi);
        t += __shfl_xor(t, 1, 16);
        t += __shfl_xor(t, 2, 16);
        t += __shfl_xor(t, 4, 16);
        t += __shfl_xor(t, 8, 16);
        if (l15 == 0) {
            const int   m  = i + 8 * grpHi;
            const float p  = t + b3v;
            const float sp = fmaxf(p, 0.0f) + log1pf(expf(-fabsf(p)));  // logaddexp(0,p)
            acc += sp - p * mask[base + m];
        }
    }
    acc += __shfl_xor(acc, 16, 32);             // combine lanes 0 and 16
    if (lane == 0) wsum[wv] = acc;
    __syncthreads();
    if (threadIdx.x == 0) {
        float s = 0.0f;
        #pragma unroll
        for (int j = 0; j < DEN_THREADS / 32; ++j) s += wsum[j];
        partial[blockIdx.x] = s;
    }
}

// ---------------- kernel 4: finalize (double accumulation) ----------------
__global__ void k_final(const float* __restrict__ aeP, const float* __restrict__ dP,
                        float* __restrict__ out) {
    __shared__ double sA[256];
    __shared__ double sD[256];
    double a = 0.0, d = 0.0;
    for (int i = threadIdx.x; i < AE_BLOCKS; i += 256) a += (double)aeP[i];
    for (int i = threadIdx.x; i < DEN_BLOCKS; i += 256) d += (double)dP[i];
    sA[threadIdx.x] = a;
    sD[threadIdx.x] = d;
    __syncthreads();
    for (int s = 128; s > 0; s >>= 1) {
        if (threadIdx.x < s) {
            sA[threadIdx.x] += sA[threadIdx.x + s];
            sD[threadIdx.x] += sD[threadIdx.x + s];
        }
        __syncthreads();
    }
    if (threadIdx.x == 0) {
        out[0] = (float)(sA[0] * (1.0 / 33554432.0));   // / (N*C*H*W) = 2^25
        out[1] = (float)(sD[0] * (1.0 / 131072.0));     // / (N*H*W)   = 2^17
    }
}

// ---------------- host launcher ----------------
extern "C" void kernel_launch(void* const* d_in, const int* in_sizes, int n_in,
                              void* d_out, int out_size, void* d_ws, size_t ws_size,
                              hipStream_t stream) {
    (void)in_sizes; (void)n_in; (void)out_size; (void)ws_size;
    const float* err        = (const float*)d_in[0];
    const float* err_map    = (const float*)d_in[1];
    const float* gt_boxes   = (const float*)d_in[2];
    const float* w1         = (const float*)d_in[3];
    const float* b1         = (const float*)d_in[4];
    const float* w2         = (const float*)d_in[5];
    const float* b2         = (const float*)d_in[6];
    const float* w3         = (const float*)d_in[7];
    const float* b3         = (const float*)d_in[8];
    const int*   gt_classes = (const int*)d_in[9];
    float*       out        = (float*)d_out;

    // workspace: [mask: NMTOT floats][aeP: 1024][dP: 1024]  (~520 KB)
    float* ws    = (float*)d_ws;
    float* maskb = ws;
    float* aeP   = ws + NMTOT;
    float* dP    = aeP + AE_BLOCKS;

    k_ae_partial<<<AE_BLOCKS, AE_THREADS, 0, stream>>>(err, aeP);
    k_mask<<<NMTOT / 256, 256, 0, stream>>>(gt_boxes, gt_classes, maskb);
    k_density<<<DEN_BLOCKS, DEN_THREADS, 0, stream>>>(err_map, w1, b1, w2, b2, w3, b3, maskb, dP);
    k_final<<<1, 256, 0, stream>>>(aeP, dP, out);
}